// TorchRnn_71055938945441
// MI455X (gfx1250) — compile-verified
//
#include <hip/hip_runtime.h>

// Vanilla tanh RNN for MI455X (gfx1250), fp32 throughout using
// V_WMMA_F32_16X16X4_F32 (native fp32 WMMA, wave32).
//
// Phase 1: xp = input @ W_ih^T  -> stored into d_out (in place; phase 2
//          overwrites with tanh output after consuming each tile).
// Phase 2: sequential scan h_t = tanh(xp_t + h_{t-1} @ W_hh^T);
//          16 WGs (one per 16 batch rows) x 8 waves (one per 16 H columns),
//          W_hh^T fragments resident in VGPRs, h double-buffered in LDS.
//
// K=128 reduction is split across 4 independent WMMA accumulator chains
// (k -> chain k&3) so consecutive v_wmma issues never depend on each other;
// the chains are summed with VALU adds that co-execute with the matrix pipe.

typedef __attribute__((ext_vector_type(2))) float v2f;
typedef __attribute__((ext_vector_type(4))) float v4f;
typedef __attribute__((ext_vector_type(8))) float v8f;

#define B_SZ 256
#define T_SZ 2048
#define I_SZ 128
#define H_SZ 128
#define LDSP 132   // padded LDS row stride (floats): 132 % 64 == 4 -> conflict-free b64 A reads

__device__ __forceinline__ float fast_tanh(float x) {
#if __has_builtin(__builtin_amdgcn_tanhf)
  return __builtin_amdgcn_tanhf(x);
#else
  // tanh(x) = sign(x) * (1 - e) / (1 + e),  e = exp(-2|x|)
  float ax = __builtin_fabsf(x);
  float e  = __builtin_amdgcn_exp2f(ax * -2.885390081777927f); // -2*log2(e)
  float t  = (1.0f - e) * __builtin_amdgcn_rcpf(1.0f + e);
  return __builtin_copysignf(t, x);
#endif
}

// ---------------- Phase 1: big parallel GEMM  xp = input @ W_ih^T ----------
__global__ __launch_bounds__(256) void rnn_xproj(const float* __restrict__ inp,
                                                 const float* __restrict__ Wih,
                                                 float* __restrict__ out) {
  __shared__ float atile[16 * LDSP];
  const int tid  = threadIdx.x;
  const int lane = tid & 31;
  const int wave = tid >> 5;          // 0..7 -> 16-col N tile
  const int cl   = lane & 15;
  const int hi   = lane >> 4;         // 0: K pair {0,1}, 1: K pair {2,3}
  const int nc0  = wave * 16;

  // B fragments = W_ih^T slice, resident in 64 VGPRs.
  // frag k, lane: B[4k + 2*hi + {0,1}][nc0+cl] = W_ih[nc0+cl][4k + 2*hi + {0,1}]
  v2f bf[32];
#pragma unroll
  for (int k = 0; k < 32; ++k)
    bf[k] = *(const v2f*)(Wih + (nc0 + cl) * I_SZ + 4 * k + 2 * hi);

  const int mt0 = blockIdx.x * 8;     // 8 M-tiles per WG, 4096 WGs total
  for (int it = 0; it < 8; ++it) {
    const long r0 = (long)(mt0 + it) * 16;       // first row of this 16x128 tile
    const float* src = inp + r0 * I_SZ;          // tile is 2048 contiguous floats
#pragma unroll
    for (int j = 0; j < 2; ++j) {
      int fi  = tid + j * 256;                   // float4 index 0..511
      int row = fi >> 5;                         // 32 float4 per row
      int c4  = fi & 31;
      v4f v = *(const v4f*)(src + (long)fi * 4);
      *(v4f*)(&atile[row * LDSP + c4 * 4]) = v;
    }
    __syncthreads();

    v8f acc[4] = {{}, {}, {}, {}};
#pragma unroll
    for (int k = 0; k < 32; ++k) {
      v2f a = *(const v2f*)(&atile[cl * LDSP + 4 * k + 2 * hi]);
      acc[k & 3] = __builtin_amdgcn_wmma_f32_16x16x4_f32(
          false, a, false, bf[k], (short)0, acc[k & 3], false, false);
    }
    v8f c = (acc[0] + acc[1]) + (acc[2] + acc[3]);
    // C/D layout: vgpr r, lanes 0-15 -> (M=r, N=lane); lanes 16-31 -> (M=r+8)
#pragma unroll
    for (int rr = 0; rr < 8; ++rr) {
      long row = r0 + rr + hi * 8;
      out[row * H_SZ + nc0 + cl] = c[rr];
    }
    __syncthreads();
  }
}

// ---------------- Phase 2: sequential recurrence ---------------------------
__global__ __launch_bounds__(256) void rnn_scan(const float* __restrict__ Whh,
                                                float* __restrict__ out) {
  __shared__ float hbuf[2][16 * LDSP];
  const int tid  = threadIdx.x;
  const int lane = tid & 31;
  const int wave = tid >> 5;
  const int cl   = lane & 15;
  const int hi   = lane >> 4;
  const int nc0  = wave * 16;
  const int m0   = blockIdx.x * 16;   // batch-row tile

  // W_hh^T fragments resident in VGPRs for the whole scan.
  v2f bf[32];
#pragma unroll
  for (int k = 0; k < 32; ++k)
    bf[k] = *(const v2f*)(Whh + (nc0 + cl) * H_SZ + 4 * k + 2 * hi);

  float h[8];
  // t = 0: h0 == 0  ->  h1 = tanh(xp_0) (skip the matmul entirely)
#pragma unroll
  for (int rr = 0; rr < 8; ++rr) {
    long row = m0 + rr + hi * 8;
    long a   = (row * T_SZ + 0) * H_SZ + nc0 + cl;
    float v  = fast_tanh(out[a]);
    out[a]   = v;
    hbuf[0][(rr + hi * 8) * LDSP + nc0 + cl] = v;
    h[rr] = v;
  }
  __syncthreads();

  for (int t = 1; t < T_SZ; ++t) {
    const int pb = (t - 1) & 1, cb = t & 1;
    // Prefetch xp_t (overlaps the LDS loads + WMMA chains below).
    float xp[8];
#pragma unroll
    for (int rr = 0; rr < 8; ++rr) {
      long row = m0 + rr + hi * 8;
      xp[rr] = out[(row * T_SZ + t) * H_SZ + nc0 + cl];
    }
    // C = h_{t-1} @ W_hh^T  (K=128 -> 32 fp32 WMMAs over 4 independent chains)
    v8f acc[4] = {{}, {}, {}, {}};
#pragma unroll
    for (int k = 0; k < 32; ++k) {
      v2f a = *(const v2f*)(&hbuf[pb][cl * LDSP + 4 * k + 2 * hi]);
      acc[k & 3] = __builtin_amdgcn_wmma_f32_16x16x4_f32(
          false, a, false, bf[k], (short)0, acc[k & 3], false, false);
    }
    v8f c = (acc[0] + acc[1]) + (acc[2] + acc[3]);
#pragma unroll
    for (int rr = 0; rr < 8; ++rr) {
      float v = fast_tanh(c[rr] + xp[rr]);
      long row = m0 + rr + hi * 8;
      out[(row * T_SZ + t) * H_SZ + nc0 + cl] = v;          // overwrite xp -> output
      hbuf[cb][(rr + hi * 8) * LDSP + nc0 + cl] = v;        // publish new h
      h[rr] = v;
    }
    __syncthreads();  // h_t visible to all waves; also guards buffer reuse
  }

  // h_n tail: [1, B, H] appended after output
  float* hn = out + (long)B_SZ * T_SZ * H_SZ;
#pragma unroll
  for (int rr = 0; rr < 8; ++rr) {
    long row = m0 + rr + hi * 8;
    hn[row * H_SZ + nc0 + cl] = h[rr];
  }
}

extern "C" void kernel_launch(void* const* d_in, const int* in_sizes, int n_in,
                              void* d_out, int out_size, void* d_ws, size_t ws_size,
                              hipStream_t stream) {
  (void)in_sizes; (void)n_in; (void)d_ws; (void)ws_size; (void)out_size;
  const float* inp = (const float*)d_in[0];   // [B, T, I]
  const float* Wih = (const float*)d_in[1];   // [H, I]
  const float* Whh = (const float*)d_in[2];   // [H, H]
  float* out = (float*)d_out;                 // [B*T*H] output ++ [B*H] h_n

  // Phase 1: 32768 M-tiles, 8 per WG -> 4096 WGs of 8 waves.
  rnn_xproj<<<(B_SZ * T_SZ) / (16 * 8), 256, 0, stream>>>(inp, Wih, out);
  // Phase 2: one WG per 16 batch rows.
  rnn_scan<<<B_SZ / 16, 256, 0, stream>>>(Whh, out);
}